// GraphConvS_9955734192544
// MI455X (gfx1250) — compile-verified
//
#include <hip/hip_runtime.h>

#define B_   64
#define M_   128
#define F1_  32
#define K_   64      // F1 + F2
#define F_   32      // FILTERS
#define LDSS 68      // padded LDS row stride (floats) -> bank-conflict-free

typedef __attribute__((ext_vector_type(2))) float v2f;
typedef __attribute__((ext_vector_type(4))) float v4f;
typedef __attribute__((ext_vector_type(8))) float v8f;

__global__ __launch_bounds__(512)
void graphconv_fused_kernel(const float* __restrict__ sf1,
                            const float* __restrict__ sf2,
                            const float* __restrict__ adj,
                            const float* __restrict__ W,
                            const float* __restrict__ bias,
                            float* __restrict__ out)
{
    __shared__ float Yt[16 * LDSS];   // 16 reduced rows of 64 (padded)
    __shared__ float degs[16];

    const int tile = blockIdx.x;           // 512 tiles of 16 rows
    const int wave = threadIdx.x >> 5;     // 0..15
    const int lane = threadIdx.x & 31;
    const int row  = tile * 16 + wave;     // row = b*M + i  in [0, 8192)

    // ---------------- Stage 1: Y[row,k] = sum_j adj[row,j] * x[row,j,k] ----
    const float* __restrict__ ar = adj + (size_t)row * M_;
    const int g = lane >> 4;               // j parity (0/1)
    const int h = (lane >> 3) & 1;         // tensor select (sf1/sf2)
    const int q = lane & 7;                // float4 chunk within 32 feats
    const float* __restrict__ xs =
        (h ? sf2 : sf1) + (size_t)row * M_ * F1_ + 4 * q;

    v4f acc = {0.f, 0.f, 0.f, 0.f};
    float da = 0.f;
    #pragma unroll 4
    for (int it = 0; it < M_ / 2; ++it) {
        const int j = 2 * it + g;
        const float a = ar[j];                                  // broadcast load
        const v4f v = *(const v4f*)(xs + (size_t)j * F1_);      // b128, coalesced
        acc += a * v;
        da  += a;
    }
    // fold the two j-parities: lane l += lane l^16
    acc.x += __shfl_xor(acc.x, 16);
    acc.y += __shfl_xor(acc.y, 16);
    acc.z += __shfl_xor(acc.z, 16);
    acc.w += __shfl_xor(acc.w, 16);
    da    += __shfl_xor(da, 16);

    if (lane < 16) {                        // lanes 0-7: sf1 k=4q, 8-15: sf2 k=32+4q
        const int k = 32 * h + 4 * q;
        *(v4f*)(&Yt[wave * LDSS + k]) = acc;
    }
    if (lane == 0) degs[wave] = da;
    __syncthreads();

    // ---------------- Stage 2: out_tile = Y_tile(16x64) @ W(64x32) + deg*b --
    if (wave < 2) {                          // wave-uniform: EXEC all ones
        const int n0   = wave * 16;          // N-tile
        const int mlo  = lane & 15;
        const int hi   = lane >> 4;          // 0 -> K 0,1 ; 1 -> K 2,3
        const int koff = hi * 2;
        const float* __restrict__ Arow = &Yt[mlo * LDSS + koff];
        const float* __restrict__ Wp   = W + n0 + mlo;

        v8f c = {0.f, 0.f, 0.f, 0.f, 0.f, 0.f, 0.f, 0.f};
        #pragma unroll
        for (int kc = 0; kc < K_; kc += 4) {
            v2f aa = *(const v2f*)(Arow + kc);             // A 16x4 layout
            v2f bb;
            bb.x = Wp[(kc + koff + 0) * F_];               // B 4x16 layout
            bb.y = Wp[(kc + koff + 1) * F_];
            c = __builtin_amdgcn_wmma_f32_16x16x4_f32(
                    /*neg_a=*/false, aa, /*neg_b=*/false, bb,
                    /*c_mod=*/(short)0, c, /*reuse_a=*/false, /*reuse_b=*/false);
        }

        // epilogue: + deg[m] * bias[n], store D (VGPR r -> M = r + hi*8)
        const float bv = bias[n0 + mlo];
        float* __restrict__ orow = out + ((size_t)tile * 16) * F_ + n0 + mlo;
        const int mbase = hi * 8;
        #pragma unroll
        for (int r = 0; r < 8; ++r) {
            const int m = mbase + r;
            orow[(size_t)m * F_] = c[r] + degs[m] * bv;
        }
    }
}

extern "C" void kernel_launch(void* const* d_in, const int* in_sizes, int n_in,
                              void* d_out, int out_size, void* d_ws, size_t ws_size,
                              hipStream_t stream) {
    (void)in_sizes; (void)n_in; (void)d_ws; (void)ws_size; (void)out_size;
    const float* sf1  = (const float*)d_in[0];
    const float* sf2  = (const float*)d_in[1];
    const float* adj  = (const float*)d_in[2];
    const float* W    = (const float*)d_in[3];
    const float* bias = (const float*)d_in[4];
    float* out = (float*)d_out;

    dim3 grid(B_ * M_ / 16);   // 512 tiles
    dim3 block(512);           // 16 waves
    hipLaunchKernelGGL(graphconv_fused_kernel, grid, block, 0, stream,
                       sf1, sf2, adj, W, bias, out);
}